// GraphEncoder_38955353375018
// MI455X (gfx1250) — compile-verified
//
#include <hip/hip_runtime.h>
#include <hip/hip_bf16.h>

// ---------------------------------------------------------------------------
// GraphSAGE 3-layer encoder for MI455X (gfx1250, wave32).
//  - Edge aggregation: wave-per-edge gather + global_atomic_add_f32 (L2-resident)
//  - SAGE linear:      V_WMMA_F32_16X16X4_F32, weights staged in LDS,
//                      deg-normalization + bias + ReLU fused.
//  - Pooling:          wave-per-node atomic scatter + divide.
// ---------------------------------------------------------------------------

#define NNODES 50000
#define NEDGES 800000
#define FDIM   128
#define NGRAPH 64
#define MTILES (NNODES / 16)   // 3125 exactly

typedef __attribute__((ext_vector_type(2))) float v2f;
typedef __attribute__((ext_vector_type(8))) float v8f;

// ------------------------------ utility ------------------------------------

__global__ void zero_f4(float* __restrict__ p, int n4) {
    int i = blockIdx.x * blockDim.x + threadIdx.x;
    if (i < n4) ((float4*)p)[i] = make_float4(0.f, 0.f, 0.f, 0.f);
}

__global__ void deg_count(const int* __restrict__ dst, float* __restrict__ deg, int E) {
    int e = blockIdx.x * blockDim.x + threadIdx.x;
    if (e < E) atomicAdd(&deg[dst[e]], 1.0f);
}

__global__ void deg_invert(float* __restrict__ deg, int n) {
    int i = blockIdx.x * blockDim.x + threadIdx.x;
    if (i < n) deg[i] = 1.0f / fmaxf(deg[i], 1.0f);
}

// --------------------------- edge aggregation ------------------------------
// One wave (32 lanes) per edge; each lane moves 4 channels (float4 = 512B/row).
__global__ void scatter_add(const int* __restrict__ src, const int* __restrict__ dst,
                            const float* __restrict__ hin, float* __restrict__ agg, int E) {
    int t = blockIdx.x * blockDim.x + threadIdx.x;
    int e = t >> 5;
    if (e >= E) return;
    int lane = t & 31;
    int s = src[e];
    int d = dst[e];
    float4 v = *(const float4*)&hin[s * FDIM + lane * 4];
    float* a = &agg[d * FDIM + lane * 4];
    atomicAdd(a + 0, v.x);
    atomicAdd(a + 1, v.y);
    atomicAdd(a + 2, v.z);
    atomicAdd(a + 3, v.w);
}

// ------------------------------ SAGE GEMM ----------------------------------
// out[m,:] = act( (agg[m,:] * dinv[m]) @ Wl^T + bl + hin[m,:] @ Wr^T )
// Block = 128 threads = 4 waves. Wave w owns output n-tiles {2w, 2w+1}.
// Weights (2 x 64KB) + bias staged in dynamic LDS (fits 320KB WGP LDS).
// NOTE: the k-loop must stay rolled (unroll<=2); a full unroll makes the LDS
// B-fragments tile-invariant straight-line code, which LICM hoists (256 VGPRs)
// and the allocator spills to scratch.
__global__ void sage_gemm(const float* __restrict__ hin,  // [N,128]
                          const float* __restrict__ agg,  // [N,128]
                          const float* __restrict__ dinv, // [N]
                          const float* __restrict__ Wl,   // [128,128] row-major [out,in]
                          const float* __restrict__ bl,   // [128]
                          const float* __restrict__ Wr,   // [128,128]
                          float* __restrict__ hout,       // [N,128]
                          int relu) {
    extern __shared__ float lds[];
    float* sWl = lds;                      // 16384 f
    float* sWr = lds + FDIM * FDIM;        // 16384 f
    float* sbl = lds + 2 * FDIM * FDIM;    // 128 f

    const int tid = threadIdx.x;           // 0..127
    for (int i = tid * 4; i < 2 * FDIM * FDIM; i += 128 * 4) {
        float4 w = *(const float4*)((i < FDIM * FDIM) ? &Wl[i] : &Wr[i - FDIM * FDIM]);
        *(float4*)&lds[i] = w;
    }
    if (tid < 32) *(float4*)&sbl[tid * 4] = *(const float4*)&bl[tid * 4];
    __syncthreads();

    const int lane = tid & 31;
    const int wave = tid >> 5;             // 0..3
    const int row  = lane & 15;            // A row within tile (M), also B/C col (N)
    const int kg   = (lane >> 4) << 1;     // K sub-group: 0 or 2
    const int n0   = (wave * 2 + 0) * 16;
    const int n1   = (wave * 2 + 1) * 16;

    // tile-invariant: bias values and LDS base pointers for this lane's columns
    const float b0 = sbl[n0 + row];
    const float b1 = sbl[n1 + row];
    const float* sWl0 = &sWl[(n0 + row) * FDIM + kg];
    const float* sWr0 = &sWr[(n0 + row) * FDIM + kg];
    const float* sWl1 = &sWl[(n1 + row) * FDIM + kg];
    const float* sWr1 = &sWr[(n1 + row) * FDIM + kg];

    for (int tile = blockIdx.x; tile < MTILES; tile += gridDim.x) {
        const int m0 = tile * 16;
        const float dv = dinv[m0 + row];
        const float* aggRow = &agg[(m0 + row) * FDIM + kg];
        const float* hinRow = &hin[(m0 + row) * FDIM + kg];

        // prefetch next tile's A rows into cache (global_prefetch_b8)
        int tn = tile + gridDim.x;
        if (tn < MTILES) {
            __builtin_prefetch(&agg[(tn * 16 + row) * FDIM], 0, 1);
            __builtin_prefetch(&hin[(tn * 16 + row) * FDIM], 0, 1);
        }

        // accumulators pre-loaded with bias (broadcast across M rows)
        v8f acc0, acc1;
#pragma unroll
        for (int j = 0; j < 8; ++j) { acc0[j] = b0; acc1[j] = b1; }

#pragma unroll 2
        for (int k0 = 0; k0 < FDIM; k0 += 4) {
            // A fragments (16x4 f32): lane = row, VGPR pair = K {kg, kg+1}
            v2f a1 = *(const v2f*)&aggRow[k0];
            a1.x *= dv; a1.y *= dv;                  // fuse mean-normalization
            v2f a2 = *(const v2f*)&hinRow[k0];

            // B fragments (4x16 f32): B[k][n] = W[n][k] -> contiguous K pair in LDS
            v2f bw0 = *(const v2f*)&sWl0[k0];
            v2f br0 = *(const v2f*)&sWr0[k0];
            v2f bw1 = *(const v2f*)&sWl1[k0];
            v2f br1 = *(const v2f*)&sWr1[k0];

            acc0 = __builtin_amdgcn_wmma_f32_16x16x4_f32(false, a1, false, bw0, (short)0, acc0, false, false);
            acc0 = __builtin_amdgcn_wmma_f32_16x16x4_f32(false, a2, false, br0, (short)0, acc0, false, false);
            acc1 = __builtin_amdgcn_wmma_f32_16x16x4_f32(false, a1, false, bw1, (short)0, acc1, false, false);
            acc1 = __builtin_amdgcn_wmma_f32_16x16x4_f32(false, a2, false, br1, (short)0, acc1, false, false);
        }

        if (relu) {
#pragma unroll
            for (int j = 0; j < 8; ++j) {
                acc0[j] = fmaxf(acc0[j], 0.f);
                acc1[j] = fmaxf(acc1[j], 0.f);
            }
        }

        // C/D layout: VGPR j holds M = j (lanes 0-15) / j+8 (lanes 16-31), N = lane&15
        const int rbase = m0 + ((lane >> 4) << 3);
#pragma unroll
        for (int j = 0; j < 8; ++j) {
            hout[(rbase + j) * FDIM + n0 + row] = acc0[j];
            hout[(rbase + j) * FDIM + n1 + row] = acc1[j];
        }
    }
}

// ------------------------------- pooling -----------------------------------

__global__ void pool_scatter(const int* __restrict__ batch, const float* __restrict__ h,
                             float* __restrict__ psum, float* __restrict__ pcnt, int N) {
    int t = blockIdx.x * blockDim.x + threadIdx.x;
    int n = t >> 5;
    if (n >= N) return;
    int lane = t & 31;
    int g = batch[n];
    float4 v = *(const float4*)&h[n * FDIM + lane * 4];
    float* p = &psum[g * FDIM + lane * 4];
    atomicAdd(p + 0, v.x);
    atomicAdd(p + 1, v.y);
    atomicAdd(p + 2, v.z);
    atomicAdd(p + 3, v.w);
    if (lane == 0) atomicAdd(&pcnt[g], 1.0f);
}

__global__ void pool_final(const float* __restrict__ psum, const float* __restrict__ pcnt,
                           float* __restrict__ out) {
    int i = blockIdx.x * blockDim.x + threadIdx.x;
    if (i < NGRAPH * FDIM) out[i] = psum[i] / fmaxf(pcnt[i >> 7], 1.0f);
}

// ------------------------------- launcher ----------------------------------

extern "C" void kernel_launch(void* const* d_in, const int* in_sizes, int n_in,
                              void* d_out, int out_size, void* d_ws, size_t ws_size,
                              hipStream_t stream) {
    const float* x   = (const float*)d_in[0];
    const int*   ei  = (const int*)d_in[1];     // [2, E]
    const int*   src = ei;
    const int*   dst = ei + NEDGES;
    const int*   bat = (const int*)d_in[2];
    const float* Wl[3] = { (const float*)d_in[3], (const float*)d_in[6], (const float*)d_in[9]  };
    const float* bl[3] = { (const float*)d_in[4], (const float*)d_in[7], (const float*)d_in[10] };
    const float* Wr[3] = { (const float*)d_in[5], (const float*)d_in[8], (const float*)d_in[11] };
    float* out = (float*)d_out;

    // workspace layout (floats)
    float* ws   = (float*)d_ws;
    float* dinv = ws;                               // 50048 (deg, then inverted in place)
    float* h0   = dinv + 50048;                     // N*128
    float* h1   = h0 + NNODES * FDIM;               // N*128
    float* agg  = h1 + NNODES * FDIM;               // N*128
    float* psum = agg + NNODES * FDIM;              // 64*128
    float* pcnt = psum + NGRAPH * FDIM;             // 64

    const int shmem = (2 * FDIM * FDIM + FDIM) * (int)sizeof(float);  // 131584 B

    // degree (structure-only, computed once)
    zero_f4<<<(NNODES / 4 + 255) / 256, 256, 0, stream>>>(dinv, NNODES / 4);
    deg_count<<<(NEDGES + 255) / 256, 256, 0, stream>>>(dst, dinv, NEDGES);
    deg_invert<<<(NNODES + 255) / 256, 256, 0, stream>>>(dinv, NNODES);

    const float* hin[3]  = { x,  h0, h1 };
    float*       hout[3] = { h0, h1, h0 };
    const int aggGrid  = ((NEDGES * 32) + 255) / 256;
    const int zeroGrid = (NNODES * FDIM / 4 + 255) / 256;

    for (int l = 0; l < 3; ++l) {
        zero_f4<<<zeroGrid, 256, 0, stream>>>(agg, NNODES * FDIM / 4);
        scatter_add<<<aggGrid, 256, 0, stream>>>(src, dst, hin[l], agg, NEDGES);
        sage_gemm<<<400, 128, shmem, stream>>>(hin[l], agg, dinv, Wl[l], bl[l], Wr[l],
                                               hout[l], (l < 2) ? 1 : 0);
    }

    // global mean pool
    zero_f4<<<((NGRAPH * FDIM + NGRAPH) / 4 + 255) / 256, 256, 0, stream>>>(psum, (NGRAPH * FDIM + NGRAPH) / 4);
    pool_scatter<<<((NNODES * 32) + 255) / 256, 256, 0, stream>>>(bat, h0, psum, pcnt, NNODES);
    pool_final<<<(NGRAPH * FDIM + 255) / 256, 256, 0, stream>>>(psum, pcnt, out);
}